// LayerNAS_Cell_38981123178600
// MI455X (gfx1250) — compile-verified
//
#include <hip/hip_runtime.h>
#include <hip/hip_bf16.h>

typedef __attribute__((ext_vector_type(2))) float v2f;
typedef __attribute__((ext_vector_type(8))) float v8f;

#define D_HID 128

// ---------------------------------------------------------------------------
// K1: init deg to 1.0 (self-loop weight), zero pooling buffers
// ---------------------------------------------------------------------------
__global__ void gcn_init(float* __restrict__ deg, float* __restrict__ sums,
                         float* __restrict__ cnt, int N, int SG, int G) {
    int i = blockIdx.x * blockDim.x + threadIdx.x;
    if (i < N)  deg[i]  = 1.0f;
    if (i < SG) sums[i] = 0.0f;
    if (i < G)  cnt[i]  = 0.0f;
}

// ---------------------------------------------------------------------------
// K2: weighted in-degree accumulation over edges
// ---------------------------------------------------------------------------
__global__ void gcn_deg(const int* __restrict__ ei, const float* __restrict__ ew,
                        float* __restrict__ deg, int E) {
    int e = blockIdx.x * blockDim.x + threadIdx.x;
    if (e >= E) return;
    atomicAdd(&deg[ei[E + e]], ew[e]);   // col = target = edge_index[1]
}

// ---------------------------------------------------------------------------
// K3: deg -> d^{-1/2} in place (deg >= 1 always due to self-loop)
// ---------------------------------------------------------------------------
__global__ void gcn_dinv(float* __restrict__ deg, int N) {
    int i = blockIdx.x * blockDim.x + threadIdx.x;
    if (i >= N) return;
    deg[i] = rsqrtf(deg[i]);
}

// ---------------------------------------------------------------------------
// K4: h = x @ W via V_WMMA_F32_16X16X4_F32. One wave -> 16x128 band of h.
// ---------------------------------------------------------------------------
__global__ __launch_bounds__(256) void gcn_gemm_wmma(const float* __restrict__ x,
                                                     const float* __restrict__ W,
                                                     float* __restrict__ h, int N) {
    const int wave    = blockIdx.x * (blockDim.x >> 5) + (threadIdx.x >> 5);
    const int lane    = threadIdx.x & 31;
    const int rowBase = wave * 16;
    if (rowBase >= N) return;                // wave-uniform; EXEC stays all-1s

    const int half = lane >> 4;              // 0: lanes 0-15, 1: lanes 16-31
    const int l16  = lane & 15;

    v8f acc[8] = {};                         // 8 N-tiles of 16 cols each

    const float* aBase = x + (size_t)(rowBase + l16) * D_HID;
    for (int k0 = 0; k0 < D_HID; k0 += 4) {
        // A 16x4 f32: lanes 0-15 carry K=k0,k0+1; lanes 16-31 carry K=k0+2,k0+3
        v2f a;
        a.x = aBase[k0 + 2 * half + 0];
        a.y = aBase[k0 + 2 * half + 1];
#pragma unroll
        for (int t = 0; t < 8; ++t) {
            // B 4x16 f32: N striped over lanes within a VGPR, K over VGPRs/halves
            const float* bp = W + (size_t)(k0 + 2 * half) * D_HID + t * 16 + l16;
            v2f bfrag;
            bfrag.x = bp[0];
            bfrag.y = bp[D_HID];
            acc[t] = __builtin_amdgcn_wmma_f32_16x16x4_f32(
                false, a, false, bfrag, (short)0, acc[t], false, false);
        }
    }
    // C/D layout: VGPR r -> row rowBase + r + 8*half, col t*16 + l16
#pragma unroll
    for (int t = 0; t < 8; ++t) {
#pragma unroll
        for (int r = 0; r < 8; ++r) {
            h[(size_t)(rowBase + r + 8 * half) * D_HID + t * 16 + l16] = acc[t][r];
        }
    }
}

// ---------------------------------------------------------------------------
// K5: out[i] = h[i] * dinv[i]^2  (self-loop message; also initializes d_out)
// ---------------------------------------------------------------------------
__global__ void gcn_selfloop(const float* __restrict__ h, const float* __restrict__ dinv,
                             float* __restrict__ out, int total) {
    int idx = blockIdx.x * blockDim.x + threadIdx.x;
    if (idx >= total) return;
    int i = idx >> 7;
    float d = dinv[i];
    out[idx] = h[idx] * d * d;
}

// ---------------------------------------------------------------------------
// K6: edge scatter: one wave per edge, 4 floats per lane, fp32 L2 atomics
// ---------------------------------------------------------------------------
__global__ void gcn_scatter(const float* __restrict__ h, const float* __restrict__ dinv,
                            const int* __restrict__ ei, const float* __restrict__ ew,
                            float* __restrict__ out, int E) {
    long long t = (long long)blockIdx.x * blockDim.x + threadIdx.x;
    long long e = t >> 5;
    int lane = (int)(t & 31);
    if (e >= E) return;
    int row = ei[e];          // source
    int col = ei[E + e];      // target
    float norm = dinv[row] * dinv[col] * ew[e];
    float4 v = ((const float4*)(h + (size_t)row * D_HID))[lane];   // coalesced 512B/edge
    float* op = out + (size_t)col * D_HID + lane * 4;
    atomicAdd(op + 0, v.x * norm);
    atomicAdd(op + 1, v.y * norm);
    atomicAdd(op + 2, v.z * norm);
    atomicAdd(op + 3, v.w * norm);
}

// ---------------------------------------------------------------------------
// K7: bias + ReLU in place, accumulate per-graph sums and counts
// ---------------------------------------------------------------------------
__global__ void gcn_relu_pool(float* __restrict__ out, const float* __restrict__ b,
                              const int* __restrict__ batch, float* __restrict__ sums,
                              float* __restrict__ cnt, int total) {
    int idx = blockIdx.x * blockDim.x + threadIdx.x;
    if (idx >= total) return;
    int i = idx >> 7;
    int k = idx & (D_HID - 1);
    float v = fmaxf(out[idx] + b[k], 0.0f);
    out[idx] = v;
    int g = batch[i];
    atomicAdd(&sums[(size_t)g * D_HID + k], v);
    if (k == 0) atomicAdd(&cnt[g], 1.0f);
}

// ---------------------------------------------------------------------------
// K8: graph embedding = sums / max(cnt, 1)
// ---------------------------------------------------------------------------
__global__ void gcn_pool_div(const float* __restrict__ sums, const float* __restrict__ cnt,
                             float* __restrict__ emb, int total) {
    int idx = blockIdx.x * blockDim.x + threadIdx.x;
    if (idx >= total) return;
    int g = idx >> 7;
    emb[idx] = sums[idx] / fmaxf(cnt[g], 1.0f);
}

// ---------------------------------------------------------------------------
extern "C" void kernel_launch(void* const* d_in, const int* in_sizes, int n_in,
                              void* d_out, int out_size, void* d_ws, size_t ws_size,
                              hipStream_t stream) {
    const float* x     = (const float*)d_in[0];
    const int*   ei    = (const int*)d_in[1];    // [2, E] int32
    const float* ew    = (const float*)d_in[2];
    const int*   batch = (const int*)d_in[3];
    const float* W     = (const float*)d_in[4];
    const float* b     = (const float*)d_in[5];

    const int N = in_sizes[0] / D_HID;           // 50000
    const int E = in_sizes[1] / 2;               // 800000
    const int G = out_size / D_HID - N;          // 500

    float* out = (float*)d_out;                  // [N, 128]
    float* emb = (float*)d_out + (size_t)N * D_HID;  // [G, 128]

    // workspace layout (floats)
    float* ws   = (float*)d_ws;
    float* deg  = ws;                                  // N (becomes dinv)
    float* h    = ws + ((N + 127) & ~127) + 128;       // N*128
    float* sums = h + (size_t)N * D_HID;               // G*128
    float* cnt  = sums + (size_t)G * D_HID;            // G

    const int B = 256;
    const int SG = G * D_HID;
    const int NH = N * D_HID;

    // K1: init
    {
        int span = (N > SG) ? N : SG;
        gcn_init<<<(span + B - 1) / B, B, 0, stream>>>(deg, sums, cnt, N, SG, G);
    }
    // K2: degree
    gcn_deg<<<(E + B - 1) / B, B, 0, stream>>>(ei, ew, deg, E);
    // K3: rsqrt
    gcn_dinv<<<(N + B - 1) / B, B, 0, stream>>>(deg, N);
    // K4: WMMA GEMM (8 waves/block, one 16-row band per wave)
    {
        int bands  = (N + 15) / 16;
        int blocks = (bands + 7) / 8;
        gcn_gemm_wmma<<<blocks, B, 0, stream>>>(x, W, h, N);
    }
    // K5: self-loop init of out
    gcn_selfloop<<<(NH + B - 1) / B, B, 0, stream>>>(h, deg, out, NH);
    // K6: edge scatter (E*32 threads)
    {
        long long tot = (long long)E * 32;
        gcn_scatter<<<(unsigned)((tot + B - 1) / B), B, 0, stream>>>(h, deg, ei, ew, out, E);
    }
    // K7: bias + ReLU + pool accumulate
    gcn_relu_pool<<<(NH + B - 1) / B, B, 0, stream>>>(out, b, batch, sums, cnt, NH);
    // K8: pool divide
    gcn_pool_div<<<(SG + B - 1) / B, B, 0, stream>>>(sums, cnt, emb, SG);
}